// VGRNN_50732153700360
// MI455X (gfx1250) — compile-verified
//
#include <hip/hip_runtime.h>

#define BB 2
#define NN 20000
#define EE 320000
#define CC 256

#define MT_PER_WAVE 4
#define WAVES_PER_BLK 8
#define MT_PER_BLOCK (MT_PER_WAVE * WAVES_PER_BLK)   // 32 M-tiles per block
#define LDSK 260                                     // padded row pitch (floats)

typedef float v2f __attribute__((ext_vector_type(2)));
typedef float v8f __attribute__((ext_vector_type(8)));

__global__ __launch_bounds__(256) void zero_f32(float* __restrict__ p, int n) {
  int i = blockIdx.x * blockDim.x + threadIdx.x;
  if (i < n) p[i] = 0.0f;
}

__global__ __launch_bounds__(256) void degree_kernel(const int* __restrict__ ei,
                                                     float* __restrict__ deg) {
  int i = blockIdx.x * blockDim.x + threadIdx.x;
  if (i >= BB * EE) return;
  int b = i / EE;
  int e = i - b * EE;
  int r = ei[(size_t)b * 2 * EE + e];               // edge_index[b,0,e]
  atomicAdd(&deg[b * NN + r], 1.0f);
}

__global__ __launch_bounds__(256) void dinv_kernel(float* __restrict__ deg) {
  int i = blockIdx.x * blockDim.x + threadIdx.x;
  if (i >= BB * NN) return;
  float d = deg[i] + 1.0f;                          // + self loop weight
  deg[i] = (d > 0.0f) ? rsqrtf(d) : 0.0f;
}

// xw[b] = x[b] @ W.  Per block: one 16-col slice of W staged in LDS (transposed,
// padded), 8 waves x 4 M-tiles each.  All tile control flow is SCALAR (readfirstlane)
// so EXEC stays all-1s for every WMMA and loads clause-group cleanly.
__global__ __launch_bounds__(256) void gemm_xw_wmma(const float* __restrict__ x,
                                                    const float* __restrict__ w,
                                                    float* __restrict__ xw) {
  __shared__ float wtile[16 * LDSK];                // [n][k], 16.6 KB

  const int tid       = threadIdx.x;
  const int waveInBlk = __builtin_amdgcn_readfirstlane(tid >> 5);  // SGPR
  const int lane      = tid & 31;

  const int tilesM  = NN / 16;                      // 1250
  const int mBlocks = (tilesM + MT_PER_BLOCK - 1) / MT_PER_BLOCK;  // 40

  int blk = blockIdx.x;                             // (b*16 + nTile)*mBlocks + mBlk
  const int mBlk  = blk % mBlocks;  blk /= mBlocks;
  const int nTile = blk % (CC / 16);
  const int b     = blk / (CC / 16);
  const int n0    = nTile * 16;

  // Cooperative, coalesced stage of W[:, n0:n0+16] into LDS (transposed).
  for (int idx = tid; idx < 16 * CC; idx += 256) {
    const int k = idx >> 4;                         // 0..255
    const int n = idx & 15;
    wtile[n * LDSK + k] = w[(size_t)k * CC + n0 + n];
  }
  __syncthreads();

  const int half = lane >> 4;                       // 0: K {0,1}; 1: K {2,3}
  const int l15  = lane & 15;

  const float* __restrict__ xb  = x  + (size_t)b * NN * CC;
  float* __restrict__       xwb = xw + (size_t)b * NN * CC;

  const int mt0 = mBlk * MT_PER_BLOCK + waveInBlk * MT_PER_WAVE;   // SGPR
  const int nmt = (tilesM - mt0 < MT_PER_WAVE) ? (tilesM - mt0) : MT_PER_WAVE;
  if (nmt <= 0) return;                             // scalar branch; no later barrier

  const float* aRow = xb + (size_t)l15 * CC + 2 * half;  // + m0*CC + k per use
  const float* bRow = &wtile[l15 * LDSK + 2 * half];     // + k per use

  if (nmt == MT_PER_WAVE) {
    // ---- fast path: no guards, 4 A-loads clause together, 4 WMMAs back-to-back
    v8f acc0 = {}, acc1 = {}, acc2 = {}, acc3 = {};
    const size_t mOff = (size_t)mt0 * 16 * CC;
    #pragma unroll 2
    for (int k = 0; k < CC; k += 4) {
      const v2f bf = *(const v2f*)(bRow + k);       // ds_load_b64
      const v2f a0 = *(const v2f*)(aRow + mOff + 0 * 16 * CC + k);
      const v2f a1 = *(const v2f*)(aRow + mOff + 1 * 16 * CC + k);
      const v2f a2 = *(const v2f*)(aRow + mOff + 2 * 16 * CC + k);
      const v2f a3 = *(const v2f*)(aRow + mOff + 3 * 16 * CC + k);
      acc0 = __builtin_amdgcn_wmma_f32_16x16x4_f32(false, a0, false, bf, (short)0, acc0, false, false);
      acc1 = __builtin_amdgcn_wmma_f32_16x16x4_f32(false, a1, false, bf, (short)0, acc1, false, false);
      acc2 = __builtin_amdgcn_wmma_f32_16x16x4_f32(false, a2, false, bf, (short)0, acc2, false, false);
      acc3 = __builtin_amdgcn_wmma_f32_16x16x4_f32(false, a3, false, bf, (short)0, acc3, false, false);
    }
    v8f accs[4] = {acc0, acc1, acc2, acc3};
    #pragma unroll
    for (int i = 0; i < MT_PER_WAVE; ++i) {
      const int m0 = (mt0 + i) * 16;
      #pragma unroll
      for (int r = 0; r < 8; ++r)                   // D: VGPR r -> M=m0+r+8*half
        xwb[(size_t)(m0 + r + 8 * half) * CC + n0 + l15] = accs[i][r];
    }
  } else {
    // ---- tail path (2 M-tiles in each of 32 waves of the grid): one tile at a time
    for (int i = 0; i < nmt; ++i) {                 // scalar loop bound
      v8f acc = {};
      const int m0 = (mt0 + i) * 16;
      #pragma unroll 4
      for (int k = 0; k < CC; k += 4) {
        const v2f bf = *(const v2f*)(bRow + k);
        const v2f a  = *(const v2f*)(aRow + (size_t)m0 * CC + k);
        acc = __builtin_amdgcn_wmma_f32_16x16x4_f32(false, a, false, bf, (short)0, acc, false, false);
      }
      #pragma unroll
      for (int r = 0; r < 8; ++r)
        xwb[(size_t)(m0 + r + 8 * half) * CC + n0 + l15] = acc[r];
    }
  }
}

// One wave per (edge or self-loop): out[row] += dinv[row]*dinv[col] * xw[col]
__global__ __launch_bounds__(256) void scatter_kernel(const int* __restrict__ ei,
                                                      const float* __restrict__ dinv,
                                                      const float* __restrict__ xw,
                                                      float* __restrict__ out) {
  const int gw   = (blockIdx.x * blockDim.x + threadIdx.x) >> 5;
  const int lane = threadIdx.x & 31;
  const int EN   = EE + NN;
  if (gw >= BB * EN) return;

  const int b = gw / EN;
  const int e = gw - b * EN;
  int r, c;
  if (e < EE) {
    r = ei[(size_t)b * 2 * EE + e];                 // edge_index[b,0,e]
    c = ei[(size_t)b * 2 * EE + EE + e];            // edge_index[b,1,e]
  } else {
    r = c = e - EE;                                 // self loop
  }
  const float norm = dinv[b * NN + r] * dinv[b * NN + c];
  const float* __restrict__ src = xw  + ((size_t)b * NN + c) * CC;
  float* __restrict__       dst = out + ((size_t)b * NN + r) * CC;

  __builtin_prefetch(src + lane * 4, 0, 0);         // global_prefetch_b8

  #pragma unroll
  for (int j = 0; j < 2; ++j) {
    const int ch = j * 128 + lane * 4;              // coalesced float4 per lane
    const float4 v = *(const float4*)(src + ch);
    atomicAdd(dst + ch + 0, norm * v.x);
    atomicAdd(dst + ch + 1, norm * v.y);
    atomicAdd(dst + ch + 2, norm * v.z);
    atomicAdd(dst + ch + 3, norm * v.w);
  }
}

__global__ __launch_bounds__(256) void relu_kernel(float* __restrict__ p, int n) {
  int i = blockIdx.x * blockDim.x + threadIdx.x;
  if (i < n) p[i] = fmaxf(p[i], 0.0f);
}

extern "C" void kernel_launch(void* const* d_in, const int* in_sizes, int n_in,
                              void* d_out, int out_size, void* d_ws, size_t ws_size,
                              hipStream_t stream) {
  const float* x  = (const float*)d_in[0];   // [B, N, C]
  const int*   ei = (const int*)d_in[1];     // [B, 2, E]
  const float* w  = (const float*)d_in[2];   // [C, C]
  float* out = (float*)d_out;                // [B, N, C]

  float* deg = (float*)d_ws;                 // B*N floats (becomes dinv in place)
  float* xw  = (float*)d_ws + (1 << 16);     // B*N*C floats at 256KB offset

  const int nOut = BB * NN * CC;             // 10,240,000
  const int nDeg = BB * NN;                  // 40,000

  zero_f32<<<(nDeg + 255) / 256, 256, 0, stream>>>(deg, nDeg);
  zero_f32<<<(nOut + 255) / 256, 256, 0, stream>>>(out, nOut);

  degree_kernel<<<(BB * EE + 255) / 256, 256, 0, stream>>>(ei, deg);
  dinv_kernel<<<(nDeg + 255) / 256, 256, 0, stream>>>(deg);

  const int tilesM  = NN / 16;                                    // 1250
  const int mBlocks = (tilesM + MT_PER_BLOCK - 1) / MT_PER_BLOCK; // 40
  const int gemmBlocks = BB * (CC / 16) * mBlocks;                // 1280
  gemm_xw_wmma<<<gemmBlocks, 256, 0, stream>>>(x, w, xw);

  const int edgeWaves = BB * (EE + NN);                           // 680,000 waves
  scatter_kernel<<<(edgeWaves * 32 + 255) / 256, 256, 0, stream>>>(ei, deg, xw, out);

  relu_kernel<<<(nOut + 255) / 256, 256, 0, stream>>>(out, nOut);
}